// RGBMem_3487513444533
// MI455X (gfx1250) — compile-verified
//
#include <hip/hip_runtime.h>
#include <hip/hip_bf16.h>

// Problem constants (match reference)
#define BSZ      16
#define NDIM     128
#define KSLOTS   65537                 // K + 1
#define NROWS    (BSZ * KSLOTS)        // 1048592 gathered rows / logits
#define NTILES   (NROWS / 16)          // 65537 (exact)
#define NDATA    1000000
#define INV_T    (1.0f / 0.07f)
#define M_EMA    0.5f

typedef __attribute__((ext_vector_type(2))) float v2f;
typedef __attribute__((ext_vector_type(4))) float v4f;
typedef __attribute__((ext_vector_type(8))) float v8f;

#if __has_builtin(__builtin_amdgcn_wmma_f32_16x16x4_f32)
#define USE_WMMA_F32X4 1
#else
#define USE_WMMA_F32X4 0
#endif

// ---------------------------------------------------------------------------
// Kernel 1: stream the 512MB memory bank into d_out (non-temporal, 128-bit)
// ---------------------------------------------------------------------------
__global__ void copy_memory_kernel(const v4f* __restrict__ src,
                                   v4f* __restrict__ dst)
{
    const size_t n4 = (size_t)NDATA * NDIM / 4;   // 32,000,000
    size_t i = (size_t)blockIdx.x * blockDim.x + threadIdx.x;
    const size_t stride = (size_t)gridDim.x * blockDim.x;
    for (; i < n4; i += stride) {
        v4f v = __builtin_nontemporal_load(&src[i]);
        __builtin_nontemporal_store(v, &dst[i]);
    }
}

// ---------------------------------------------------------------------------
// Kernel 2: EMA update + L2 renorm of the 16 positive rows, plus labels = 0.
// One wave32 per batch row; 4 floats per lane; shfl_xor norm reduction.
// Must run AFTER copy_memory_kernel (same stream => ordered).
// ---------------------------------------------------------------------------
__global__ void ema_update_kernel(const float* __restrict__ x,
                                  const int*   __restrict__ y,
                                  const float* __restrict__ mem,
                                  float* __restrict__ out_mem,
                                  float* __restrict__ labels)
{
    const int lane = threadIdx.x & 31;
    const int wv   = threadIdx.x >> 5;           // 0..15 -> batch row

    if (threadIdx.x < BSZ) labels[threadIdx.x] = 0.0f;
    if (wv >= BSZ) return;

    const int row = y[wv];
    const v4f m4 = *(const v4f*)(mem + (size_t)row * NDIM + lane * 4);
    const v4f x4 = *(const v4f*)(x + wv * NDIM + lane * 4);

    v4f w4;
    w4.x = M_EMA * m4.x + (1.0f - M_EMA) * x4.x;
    w4.y = M_EMA * m4.y + (1.0f - M_EMA) * x4.y;
    w4.z = M_EMA * m4.z + (1.0f - M_EMA) * x4.z;
    w4.w = M_EMA * m4.w + (1.0f - M_EMA) * x4.w;

    float ss = w4.x * w4.x + w4.y * w4.y + w4.z * w4.z + w4.w * w4.w;
#pragma unroll
    for (int m = 16; m >= 1; m >>= 1) ss += __shfl_xor(ss, m, 32);

    const float scale = 1.0f / fmaxf(sqrtf(ss), 1e-12f);
    v4f o;
    o.x = w4.x * scale; o.y = w4.y * scale; o.z = w4.z * scale; o.w = w4.w * scale;
    *(v4f*)(out_mem + (size_t)row * NDIM + lane * 4) = o;
}

// ---------------------------------------------------------------------------
// Kernel 3: logits via V_WMMA_F32_16X16X4_F32.
// Per wave tile of 16 flattened (b,k) rows:
//   A (16x128, gathered rows) streamed via LDS, B = X^T (128x16) in registers,
//   D[m][n] = dot(row_m, x_n); extract column b_m per row m.
// ---------------------------------------------------------------------------
#define LOG_WAVES 4            // waves per block
#define LDS_ROW   132          // padded row stride (floats): 132 % 64 == 4
                               //  -> strided b64 fragment reads are bank-conflict-free

__global__ __launch_bounds__(LOG_WAVES * 32) void
logits_wmma_kernel(const float* __restrict__ x,
                   const int*   __restrict__ y,
                   const int*   __restrict__ idx,
                   const float* __restrict__ mem,
                   float* __restrict__ logits)
{
    __shared__ float ldsA[LOG_WAVES][16 * LDS_ROW];   // gathered A tiles
    __shared__ float ldsD[LOG_WAVES][16 * 16];        // D staging for extraction

    const int lane = threadIdx.x & 31;
    const int wv   = threadIdx.x >> 5;
    const int half = lane >> 4;      // 0: lanes 0-15, 1: lanes 16-31
    const int lm   = lane & 15;      // row (A/D) or column (B) within half-wave

#if USE_WMMA_F32X4
    // B fragments for all 32 K-chunks: B[k][n] = x[n][4c+k].
    // Lane layout (4x16 f32 B): n = lane%16, v0/v1 hold K = 4c + 2*half + {0,1}.
    v2f bfr[32];
#pragma unroll
    for (int c = 0; c < 32; ++c) {
        const float* px = x + lm * NDIM + 4 * c + 2 * half;
        bfr[c].x = px[0];
        bfr[c].y = px[1];
    }
#endif

    float* A = ldsA[wv];
    float* D = ldsD[wv];

    const int waves_total = gridDim.x * (blockDim.x >> 5);
    const int gw = blockIdx.x * (blockDim.x >> 5) + wv;

    for (int tile = gw; tile < NTILES; tile += waves_total) {
        // This lane's row of the tile (lanes 16-31 mirror 0-15).
        const int r  = tile * 16 + lm;
        const int b  = r / KSLOTS;
        const int k  = r - b * KSLOTS;
        const int id = (k == 0) ? y[b] : idx[r];      // slot 0 = positive key y[b]

        // Cooperative gather: one full 512B memory row per wave instruction.
        for (int j = 0; j < 16; ++j) {
            const int idj = __builtin_amdgcn_readlane(id, j);   // uniform row id
            const v4f v = *(const v4f*)(mem + (size_t)idj * NDIM + lane * 4);
            *(v4f*)(A + j * LDS_ROW + lane * 4) = v;
        }
        __builtin_amdgcn_wave_barrier();

        v8f acc = {};
#if USE_WMMA_F32X4
#pragma unroll
        for (int c = 0; c < 32; ++c) {
            // A fragment (16x4 f32): row m = lane%16, K = 4c + 2*half + {0,1}
            const float* pa = A + lm * LDS_ROW + 4 * c + 2 * half;
            v2f afr;
            afr.x = pa[0];
            afr.y = pa[1];
            acc = __builtin_amdgcn_wmma_f32_16x16x4_f32(
                false, afr, false, bfr[c], (short)0, acc, false, false);
        }
#else
        // FMA fallback with identical C/D register layout (lane: N=lm, M=j+8*half)
#pragma unroll 4
        for (int c = 0; c < 32; ++c) {
            const float* pb = x + lm * NDIM + 4 * c;
#pragma unroll
            for (int j = 0; j < 8; ++j) {
                const float* pa = A + (j + 8 * half) * LDS_ROW + 4 * c;
                acc[j] += pa[0] * pb[0] + pa[1] * pb[1] + pa[2] * pb[2] + pa[3] * pb[3];
            }
        }
#endif

        // Stage D (component j of this lane is D[M = j + 8*half][N = lm]).
#pragma unroll
        for (int j = 0; j < 8; ++j)
            D[(j + 8 * half) * 16 + lm] = acc[j];
        __builtin_amdgcn_wave_barrier();

        // Row m only needs column b_m; lanes 0-15 emit a coalesced 64B store.
        if (lane < 16)
            logits[r] = D[lm * 16 + b] * INV_T;
        __builtin_amdgcn_wave_barrier();   // keep D reads ahead of next tile's writes
    }
}

// ---------------------------------------------------------------------------
// Launch: d_out = [logits (16x65537) | labels (16) | new_memory (1e6 x 128)]
// ---------------------------------------------------------------------------
extern "C" void kernel_launch(void* const* d_in, const int* in_sizes, int n_in,
                              void* d_out, int out_size, void* d_ws, size_t ws_size,
                              hipStream_t stream)
{
    const float* x   = (const float*)d_in[0];
    const int*   y   = (const int*)d_in[1];
    const int*   idx = (const int*)d_in[2];
    const float* mem = (const float*)d_in[3];

    float* logits  = (float*)d_out;
    float* labels  = logits + NROWS;
    float* out_mem = labels + BSZ;

    // 512MB streaming copy of the memory bank into the output.
    copy_memory_kernel<<<16384, 256, 0, stream>>>((const v4f*)mem, (v4f*)out_mem);
    // Overwrite the 16 EMA-updated rows (ordered after the copy on this stream).
    ema_update_kernel<<<1, BSZ * 32, 0, stream>>>(x, y, mem, out_mem, labels);
    // Gather + WMMA logits (independent of the copy).
    logits_wmma_kernel<<<4096, LOG_WAVES * 32, 0, stream>>>(x, y, idx, mem, logits);
}